// GAT_5282809775004
// MI455X (gfx1250) — compile-verified
//
#include <hip/hip_runtime.h>
#include <hip/hip_bf16.h>
#include <float.h>

// ---------------------------------------------------------------------------
// GAT 3-layer forward for MI455X (gfx1250, wave32).
// GEMMs use V_WMMA_F32_16X16X4_F32 (full fp32 precision; GEMM cost ~1.6 GFLOP
// total, negligible). Edge softmax + scatter-aggregate dominate; working set
// (~45 MB) fits in the 192 MB L2, so edge kernels are written for maximum
// parallelism (one thread per edge-feature) to saturate L2 atomic throughput.
// ---------------------------------------------------------------------------

typedef __attribute__((ext_vector_type(2))) float v2f;
typedef __attribute__((ext_vector_type(8))) float v8f;

#define HID 64
#define NEG_SLOPE 0.2f

// ---- order-preserving float <-> uint mapping for atomic max -----------------
__device__ __forceinline__ unsigned f2ord(float f) {
  unsigned u = __float_as_uint(f);
  return (u & 0x80000000u) ? ~u : (u | 0x80000000u);
}
__device__ __forceinline__ float ord2f(unsigned u) {
  unsigned v = (u & 0x80000000u) ? (u & 0x7fffffffu) : ~u;
  return __uint_as_float(v);
}

// ---- edge (src,dst) with implicit self-loops appended ----------------------
__device__ __forceinline__ void edge_sd(const int* __restrict__ ei, int E,
                                        int e, int& s, int& d) {
  if (e < E) { s = ei[e]; d = ei[E + e]; }
  else       { s = e - E; d = e - E; }
}

// ---------------------------------------------------------------------------
// GEMM: H[nrows x 64] = X[nrows x DIN] * W[DIN x 64]  via v_wmma_f32_16x16x4_f32
// One wave per 16-row tile; 4 accumulators cover the 64 output columns.
// A-row index is CLAMPED (not predicated) so the K-loop has no divergent
// branches: loads of the duplicated clamp row are harmless because the
// corresponding output rows are never stored.
// ---------------------------------------------------------------------------
template <int DIN>
__global__ __launch_bounds__(128)
void k_gemm64(const float* __restrict__ X, const float* __restrict__ W,
              float* __restrict__ H, int nrows) {
  const int wave = threadIdx.x >> 5;
  const int lane = threadIdx.x & 31;
  const int tile = blockIdx.x * 4 + wave;
  const int m0 = tile * 16;
  if (m0 >= nrows) return;             // wave-uniform: EXEC stays all-ones

  const int half = lane >> 4;          // 0 for lanes 0-15, 1 for lanes 16-31
  const int lm   = lane & 15;
  const int row  = m0 + lm;            // A-matrix row this lane supplies
  const int lrow = (row < nrows) ? row : (nrows - 1);  // clamp, no branch
  const float* __restrict__ xr = X + (size_t)lrow * DIN;

  v8f acc0 = {}, acc1 = {}, acc2 = {}, acc3 = {};

#pragma unroll
  for (int k0 = 0; k0 < DIN; k0 += 4) {
    const int kb = k0 + half * 2;      // K pair this half-wave holds
    const v2f a = *(const v2f*)(xr + kb);               // 8B-aligned b64 load

    const float* w0 = W + (size_t)kb * HID + lm;        // row kb   of W
    const float* w1 = W + (size_t)(kb + 1) * HID + lm;  // row kb+1 of W
    v2f b0, b1, b2, b3;
    b0.x = w0[0];  b0.y = w1[0];
    b1.x = w0[16]; b1.y = w1[16];
    b2.x = w0[32]; b2.y = w1[32];
    b3.x = w0[48]; b3.y = w1[48];

    acc0 = __builtin_amdgcn_wmma_f32_16x16x4_f32(false, a, false, b0, (short)0, acc0, false, false);
    acc1 = __builtin_amdgcn_wmma_f32_16x16x4_f32(false, a, false, b1, (short)0, acc1, false, false);
    acc2 = __builtin_amdgcn_wmma_f32_16x16x4_f32(false, a, false, b2, (short)0, acc2, false, false);
    acc3 = __builtin_amdgcn_wmma_f32_16x16x4_f32(false, a, false, b3, (short)0, acc3, false, false);
  }

  const int orow = m0 + half * 8;      // C/D layout: reg r -> row orow + r
#pragma unroll
  for (int r = 0; r < 8; ++r) {
    const int rr = orow + r;
    if (rr < nrows) {
      float* o = H + (size_t)rr * HID + lm;
      o[0]  = acc0[r];
      o[16] = acc1[r];
      o[32] = acc2[r];
      o[48] = acc3[r];
    }
  }
}

// ---- per-node attention scalars: as[i] = h_i . a_src ; ad[i] = h_i . a_dst --
__global__ void k_alpha(const float* __restrict__ H,
                        const float* __restrict__ av_s,
                        const float* __restrict__ av_d,
                        float* __restrict__ as_, float* __restrict__ ad_, int n) {
  int i = blockIdx.x * blockDim.x + threadIdx.x;
  if (i >= n) return;
  const float* h = H + (size_t)i * HID;
  float s = 0.0f, d = 0.0f;
#pragma unroll
  for (int j = 0; j < HID; ++j) {
    float v = h[j];
    s = fmaf(v, av_s[j], s);
    d = fmaf(v, av_d[j], d);
  }
  as_[i] = s;
  ad_[i] = d;
}

__global__ void k_init_nodes(unsigned* __restrict__ m, float* __restrict__ s, int n) {
  int i = blockIdx.x * blockDim.x + threadIdx.x;
  if (i >= n) return;
  m[i] = f2ord(-FLT_MAX);
  s[i] = 0.0f;
}

__global__ void k_zero(float* __restrict__ p, int n) {
  int i = blockIdx.x * blockDim.x + threadIdx.x;
  if (i < n) p[i] = 0.0f;
}

// ---- edge phase 1: leaky-relu score + segment max (ordered-uint atomicMax) --
__global__ void k_edge_score_max(const int* __restrict__ ei, int E, int n,
                                 const float* __restrict__ as_,
                                 const float* __restrict__ ad_,
                                 float* __restrict__ eb, unsigned* __restrict__ m) {
  int e = blockIdx.x * blockDim.x + threadIdx.x;
  if (e >= E + n) return;
  int s, d;
  edge_sd(ei, E, e, s, d);
  float v = as_[s] + ad_[d];
  v = (v >= 0.0f) ? v : NEG_SLOPE * v;
  eb[e] = v;
  atomicMax(m + d, f2ord(v));
}

// ---- edge phase 2: p = exp(e - m[dst]); segment sum ------------------------
__global__ void k_edge_exp_sum(const int* __restrict__ ei, int E, int n,
                               const unsigned* __restrict__ m,
                               float* __restrict__ eb, float* __restrict__ sb) {
  int e = blockIdx.x * blockDim.x + threadIdx.x;
  if (e >= E + n) return;
  int d = (e < E) ? ei[E + e] : (e - E);
  float mv = ord2f(m[d]);
  if (!(mv > -FLT_MAX)) mv = 0.0f;     // mirrors the isfinite guard
  float p = expf(eb[e] - mv);
  eb[e] = p;
  atomicAdd(sb + d, p);
}

// ---- edge phase 3: out[dst] += (p/s[dst]) * h[src], one thread per (e,f) ---
__global__ void k_edge_agg(const int* __restrict__ ei, int E, int n,
                           const float* __restrict__ pb,
                           const float* __restrict__ sb,
                           const float* __restrict__ H,
                           float* __restrict__ out) {
  long long idx = (long long)blockIdx.x * blockDim.x + threadIdx.x;
  int e = (int)(idx >> 6);
  int f = (int)(idx & 63);
  if (e >= E + n) return;
  int s, d;
  edge_sd(ei, E, e, s, d);
  float w = pb[e] / (sb[d] + 1e-16f);
  atomicAdd(out + (size_t)d * HID + f, w * H[(size_t)s * HID + f]);
}

__global__ void k_relu(float* __restrict__ x, int n) {
  int i = blockIdx.x * blockDim.x + threadIdx.x;
  if (i < n) x[i] = fmaxf(x[i], 0.0f);
}

__global__ void k_log_softmax(const float* __restrict__ in,
                              float* __restrict__ out, int n) {
  int i = blockIdx.x * blockDim.x + threadIdx.x;
  if (i >= n) return;
  const float* r = in + (size_t)i * HID;
  float m = -FLT_MAX;
#pragma unroll
  for (int j = 0; j < HID; ++j) m = fmaxf(m, r[j]);
  float s = 0.0f;
#pragma unroll
  for (int j = 0; j < HID; ++j) s += expf(r[j] - m);
  float lse = m + logf(s);
  float* o = out + (size_t)i * HID;
#pragma unroll
  for (int j = 0; j < HID; ++j) o[j] = r[j] - lse;
}

// ---------------------------------------------------------------------------
static inline unsigned cdiv(long long a, long long b) { return (unsigned)((a + b - 1) / b); }

extern "C" void kernel_launch(void* const* d_in, const int* in_sizes, int n_in,
                              void* d_out, int out_size, void* d_ws, size_t ws_size,
                              hipStream_t stream) {
  const float* x  = (const float*)d_in[0];
  const int*   ei = (const int*)d_in[1];
  const int N_ = in_sizes[0] / 128;    // IN = 128
  const int E_ = in_sizes[1] / 2;
  const int EN = E_ + N_;
  const size_t N64 = (size_t)N_ * HID;

  // workspace layout (all fp32-sized)
  float*    bufA = (float*)d_ws;        // N*64  (layer io / agg output)
  float*    bufB = bufA + N64;          // N*64  (gemm output h)
  float*    as_  = bufB + N64;          // N
  float*    ad_  = as_ + N_;            // N
  unsigned* mo   = (unsigned*)(ad_ + N_); // N (ordered max)
  float*    sb   = (float*)(mo + N_);   // N (softmax denom)
  float*    eb   = sb + N_;             // E+N (scores -> probs)

  const dim3 B256(256);

  for (int l = 0; l < 3; ++l) {
    const float* X    = (l == 0) ? x : bufA;
    const float* W    = (const float*)d_in[2 + 3 * l];
    const float* avS  = (const float*)d_in[3 + 3 * l];
    const float* avD  = (const float*)d_in[4 + 3 * l];

    const unsigned gemm_blocks = cdiv(cdiv(N_, 16), 4);
    if (l == 0)
      k_gemm64<128><<<gemm_blocks, 128, 0, stream>>>(X, W, bufB, N_);
    else
      k_gemm64<64><<<gemm_blocks, 128, 0, stream>>>(X, W, bufB, N_);

    k_alpha<<<cdiv(N_, 256), B256, 0, stream>>>(bufB, avS, avD, as_, ad_, N_);
    k_init_nodes<<<cdiv(N_, 256), B256, 0, stream>>>(mo, sb, N_);
    k_zero<<<cdiv((long long)N64, 256), B256, 0, stream>>>(bufA, (int)N64);

    k_edge_score_max<<<cdiv(EN, 256), B256, 0, stream>>>(ei, E_, N_, as_, ad_, eb, mo);
    k_edge_exp_sum<<<cdiv(EN, 256), B256, 0, stream>>>(ei, E_, N_, mo, eb, sb);
    k_edge_agg<<<cdiv((long long)EN * HID, 256), B256, 0, stream>>>(ei, E_, N_, eb, sb, bufB, bufA);

    if (l < 2) k_relu<<<cdiv((long long)N64, 256), B256, 0, stream>>>(bufA, (int)N64);
  }

  k_log_softmax<<<cdiv(N_, 256), B256, 0, stream>>>(bufA, (float*)d_out, N_);
}